// RowCrossAttention_38122129719946
// MI455X (gfx1250) — compile-verified
//
#include <hip/hip_runtime.h>
#include <math.h>

typedef __attribute__((ext_vector_type(2))) float v2f;
typedef __attribute__((ext_vector_type(4))) float v4f;
typedef __attribute__((ext_vector_type(8))) float v8f;

#define R_ 64
#define C_ 1024
#define B_ 2
#define E_ 256
#define H_ 8
#define D_ 32
#define TOKENS (R_*C_*B_)
#define SCALING 0.125f     // (D^-0.5)/sqrt(B) = (1/sqrt(32))/sqrt(2)

// LDS byte address of a __shared__ object (generic -> addrspace(3) cast).
static __device__ __forceinline__ unsigned lds_addr(const void* p) {
  return (unsigned)(unsigned long long)
         (__attribute__((address_space(3))) const char*)p;
}

// ---------------------------------------------------------------------------
// Kernel 1 (tiny, 1 block): q projection + fold q into Wk/bk:
//   qs[n][e'] = (query @ Wq^T + bq)[n,e'] * SCALING
//   u[n,h,e]  = sum_d qs[n,h*32+d] * Wk[h*32+d, e]
//   t[n,h]    = R * sum_d qs[n,h*32+d] * bk[h*32+d]
// ---------------------------------------------------------------------------
__global__ __launch_bounds__(256) void qu_kernel(
    const float* __restrict__ query, const float* __restrict__ Wq,
    const float* __restrict__ bq,    const float* __restrict__ Wk,
    const float* __restrict__ bk,
    float* __restrict__ u, float* __restrict__ t)
{
  __shared__ float qs[B_][E_];
  const int tid = threadIdx.x;
  for (int n = 0; n < B_; ++n) {
    float acc = bq[tid];
    for (int q = 0; q < E_; ++q) acc += query[n*E_ + q] * Wq[tid*E_ + q];
    qs[n][tid] = acc * SCALING;
  }
  __syncthreads();
  for (int nh = 0; nh < B_*H_; ++nh) {
    const int n = nh >> 3, h = nh & 7;
    float acc = 0.f;
    for (int d = 0; d < D_; ++d) acc += qs[n][h*D_ + d] * Wk[(h*D_ + d)*E_ + tid];
    u[nh*E_ + tid] = acc;
  }
  if (tid < B_*H_) {
    const int n = tid >> 3, h = tid & 7;
    float acc = 0.f;
    for (int d = 0; d < D_; ++d) acc += qs[n][h*D_ + d] * bk[h*D_ + d];
    t[tid] = (float)R_ * acc;
  }
}

// ---------------------------------------------------------------------------
// Kernel 2: per (j,n) column: reduce key over r (the only read of `key`),
// then dot with u[] to produce attn_weights[h,n,j].  b128 loads, 4-way
// r-parallel partial sums reduced through LDS.
// ---------------------------------------------------------------------------
__global__ __launch_bounds__(256) void score_kernel(
    const float* __restrict__ key, const float* __restrict__ u,
    const float* __restrict__ t,   float* __restrict__ attn)
{
  __shared__ float sks[E_];
  __shared__ v4f  part[4][64];
  const int tid = threadIdx.x;
  const int q   = tid >> 6;             // r-quarter (0..3)
  const int e4  = tid & 63;             // float4 column
  const int j = blockIdx.x >> 1;
  const int n = blockIdx.x & 1;
  const v4f* p = (const v4f*)(key + (j*B_ + n)*E_) + e4;
  const int rs4 = C_*B_*E_/4;           // r stride in v4f units
  v4f acc = {};
#pragma unroll 4
  for (int r = q*16; r < q*16 + 16; ++r) {
    if (r + 1 < q*16 + 16) __builtin_prefetch(&p[(r + 1)*rs4], 0, 1);
    acc += p[r*rs4];
  }
  part[q][e4] = acc;
  __syncthreads();
  if (tid < 64) {
    v4f s = (part[0][tid] + part[1][tid]) + (part[2][tid] + part[3][tid]);
    *(v4f*)&sks[tid*4] = s;
  }
  __syncthreads();
  if (tid < H_) {
    float acc2 = t[n*H_ + tid];
    const float* ur = u + (n*H_ + tid)*E_;
    for (int ee = 0; ee < E_; ++ee) acc2 += ur[ee] * sks[ee];
    attn[(tid*B_ + n)*C_ + j] = acc2;
  }
}

// ---------------------------------------------------------------------------
// Kernel 3: softmax over C for each of the H*B = 16 rows -> probs (in d_out)
// ---------------------------------------------------------------------------
__global__ __launch_bounds__(256) void softmax_kernel(
    const float* __restrict__ attn, float* __restrict__ probs)
{
  __shared__ float red[256];
  const int tid = threadIdx.x;
  const int row = blockIdx.x;          // h*B + n
  const float* x = attn + row*C_;
  float v[4];
  float m = -1e30f;
  for (int i = 0; i < 4; ++i) { v[i] = x[tid + i*256]; m = fmaxf(m, v[i]); }
  red[tid] = m; __syncthreads();
  for (int s = 128; s > 0; s >>= 1) {
    if (tid < s) red[tid] = fmaxf(red[tid], red[tid + s]);
    __syncthreads();
  }
  m = red[0]; __syncthreads();
  float sum = 0.f;
  for (int i = 0; i < 4; ++i) { v[i] = __expf(v[i] - m); sum += v[i]; }
  red[tid] = sum; __syncthreads();
  for (int s = 128; s > 0; s >>= 1) {
    if (tid < s) red[tid] += red[tid + s];
    __syncthreads();
  }
  const float inv = 1.f / red[0];
  float* o = probs + row*C_;
  for (int i = 0; i < 4; ++i) o[tid + i*256] = v[i] * inv;
}

// ---------------------------------------------------------------------------
// Kernel 4: fused  out = ((value @ Wv^T + bv) * probs) @ Wo^T + bo
// 64-token tile per block.  Each of the 8 waves owns a disjoint 32-column
// slice across all four 16-row tiles -> every weight element hits L2 exactly
// once per block.  Value tile arrives via GLOBAL_LOAD_ASYNC_TO_LDS_B128
// (ASYNCcnt path, no VGPR staging).  Intermediate V' staged in padded LDS.
// ---------------------------------------------------------------------------
#define TILE_M 64
#define LDP (E_ + 4)   // padded LDS row stride (260 floats) -> conflict-free

__global__ __launch_bounds__(256) void fused_vo_kernel(
    const float* __restrict__ value,
    const float* __restrict__ Wv, const float* __restrict__ bv,
    const float* __restrict__ Wo, const float* __restrict__ bo,
    const float* __restrict__ probs,
    float* __restrict__ out)
{
  __shared__ float Xs[TILE_M * LDP];   // value tile, later re-used for V'
  __shared__ float Ps[TILE_M][H_];     // per-token per-head probs
  const int tid  = threadIdx.x;
  const int lane = tid & 31;
  const int wave = tid >> 5;
  const int tileBase = blockIdx.x * TILE_M;

  // ---- async copy: 64 value rows (64 KB) straight into LDS --------------
  {
    const char* gbase = (const char*)(value + (size_t)tileBase * E_);
#pragma unroll
    for (int it = 0; it < 16; ++it) {
      const int i  = tid + it * 256;        // b128 chunk index (4096 total)
      const int m  = i >> 6;                // row (64 chunks per 256-f row)
      const int c  = (i & 63) << 2;         // column (floats)
      const unsigned laddr = lds_addr(&Xs[m*LDP + c]);
      const unsigned voff  = (unsigned)(i << 4);   // global byte offset
      asm volatile("global_load_async_to_lds_b128 %0, %1, %2"
                   :: "v"(laddr), "v"(voff), "s"(gbase) : "memory");
    }
  }
  for (int i = tid; i < TILE_M*H_; i += 256) {
    const int m = i >> 3, h = i & 7;
    const int tok = tileBase + m;
    const int n = tok & (B_ - 1);
    const int j = (tok >> 1) & (C_ - 1);
    Ps[m][h] = probs[(h*B_ + n)*C_ + j];
  }
  asm volatile("s_wait_asynccnt 0x0" ::: "memory");
  __syncthreads();

  const int rowSel = lane & 15;
  const int kSel   = (lane >> 4) << 1;   // lanes 0-15 -> K{0,1}, 16-31 -> K{2,3}
  const int nBase  = wave * 32;          // disjoint per wave

  // ---- Stage 1: V = X @ Wv^T --------------------------------------------
  v8f acc[4][2] = {};
  for (int k0 = 0; k0 < E_; k0 += 4) {
    v2f a[4];
#pragma unroll
    for (int mT = 0; mT < 4; ++mT) {
      a[mT].x = Xs[(mT*16 + rowSel)*LDP + k0 + kSel];
      a[mT].y = Xs[(mT*16 + rowSel)*LDP + k0 + kSel + 1];
    }
    v2f b[2];
#pragma unroll
    for (int nt = 0; nt < 2; ++nt)
      b[nt] = *(const v2f*)(Wv + (nBase + nt*16 + rowSel)*E_ + k0 + kSel);
#pragma unroll
    for (int mT = 0; mT < 4; ++mT)
#pragma unroll
      for (int nt = 0; nt < 2; ++nt)
        acc[mT][nt] = __builtin_amdgcn_wmma_f32_16x16x4_f32(
            false, a[mT], false, b[nt], (short)0, acc[mT][nt], false, false);
  }
  __syncthreads();   // all waves done reading Xs before overwrite

  // epilogue: + bv, scale by per-head prob, write V' back into Xs
#pragma unroll
  for (int nt = 0; nt < 2; ++nt) {
    const int n = nBase + nt*16 + rowSel;
    const float bvn = bv[n];
    const int h = n >> 5;                   // head = e'/32
#pragma unroll
    for (int mT = 0; mT < 4; ++mT)
#pragma unroll
      for (int i = 0; i < 8; ++i) {
        const int m = mT*16 + i + ((lane >> 4) << 3);   // WMMA C layout
        Xs[m*LDP + n] = (acc[mT][nt][i] + bvn) * Ps[m][h];
      }
  }
  __syncthreads();

  // ---- Stage 2: OUT = V' @ Wo^T + bo ------------------------------------
  v8f acc2[4][2] = {};
  for (int k0 = 0; k0 < E_; k0 += 4) {
    v2f a[4];
#pragma unroll
    for (int mT = 0; mT < 4; ++mT) {
      a[mT].x = Xs[(mT*16 + rowSel)*LDP + k0 + kSel];
      a[mT].y = Xs[(mT*16 + rowSel)*LDP + k0 + kSel + 1];
    }
    v2f b[2];
#pragma unroll
    for (int nt = 0; nt < 2; ++nt)
      b[nt] = *(const v2f*)(Wo + (nBase + nt*16 + rowSel)*E_ + k0 + kSel);
#pragma unroll
    for (int mT = 0; mT < 4; ++mT)
#pragma unroll
      for (int nt = 0; nt < 2; ++nt)
        acc2[mT][nt] = __builtin_amdgcn_wmma_f32_16x16x4_f32(
            false, a[mT], false, b[nt], (short)0, acc2[mT][nt], false, false);
  }
  float* dst = out + (size_t)tileBase * E_;
#pragma unroll
  for (int nt = 0; nt < 2; ++nt) {
    const int n = nBase + nt*16 + rowSel;
    const float bon = bo[n];
#pragma unroll
    for (int mT = 0; mT < 4; ++mT)
#pragma unroll
      for (int i = 0; i < 8; ++i) {
        const int m = mT*16 + i + ((lane >> 4) << 3);
        dst[m*E_ + n] = acc2[mT][nt][i] + bon;
      }
  }
}

// ---------------------------------------------------------------------------
extern "C" void kernel_launch(void* const* d_in, const int* in_sizes, int n_in,
                              void* d_out, int out_size, void* d_ws, size_t ws_size,
                              hipStream_t stream) {
  const float* query = (const float*)d_in[0];
  const float* key   = (const float*)d_in[1];
  const float* value = (const float*)d_in[2];
  const float* Wq    = (const float*)d_in[3];
  const float* bq    = (const float*)d_in[4];
  const float* Wk    = (const float*)d_in[5];
  const float* bk    = (const float*)d_in[6];
  const float* Wv    = (const float*)d_in[7];
  const float* bv    = (const float*)d_in[8];
  const float* Wo    = (const float*)d_in[9];
  const float* bo    = (const float*)d_in[10];

  float* out = (float*)d_out;
  float* ws  = (float*)d_ws;
  float* u    = ws;                    // [B,H,E]  = 4096 floats
  float* t    = ws + 4096;             // [B,H]    = 16 floats
  float* attn = ws + 4096 + 16;        // [H,B,C]  = 16384 floats
  float* probs = out + (size_t)TOKENS * E_;  // attn_probs lives in d_out tail

  qu_kernel<<<dim3(1), dim3(256), 0, stream>>>(query, Wq, bq, Wk, bk, u, t);
  score_kernel<<<dim3(C_*B_), dim3(256), 0, stream>>>(key, u, t, attn);
  softmax_kernel<<<dim3(H_*B_), dim3(256), 0, stream>>>(attn, probs);
  fused_vo_kernel<<<dim3(TOKENS/TILE_M), dim3(256), 0, stream>>>(
      value, Wv, bv, Wo, bo, probs, out);
}